// Functional_58128087384930
// MI455X (gfx1250) — compile-verified
//
#include <hip/hip_runtime.h>
#include <math.h>

// ---------------- problem constants ----------------
#define NZ      262144            // grid length (2^18 = 4^9)
#define NQ      (NZ / 4)          // 65536 butterflies per row per radix-4 pass
#define NG      (NZ / 2 + 1)      // rfft bins
#define BATCH   8
#define NPASS   9                 // log4(NZ)
#define TWO_PI  6.28318530717958647692f
#define DZ_F    0.05f

typedef __attribute__((ext_vector_type(2))) float v2f;
typedef __attribute__((ext_vector_type(8))) float v8f;

// ---------------- complex helpers ----------------
__device__ __forceinline__ float2 cadd(float2 a, float2 b) { return make_float2(a.x + b.x, a.y + b.y); }
__device__ __forceinline__ float2 csub(float2 a, float2 b) { return make_float2(a.x - b.x, a.y - b.y); }
__device__ __forceinline__ float2 crot(float2 v, float c, float s) {
    return make_float2(v.x * c - v.y * s, v.x * s + v.y * c);
}

// ================= FFT kernels (radix-4 Stockham, auto-sort, ping-pong) ============
// Pass math (Govindaraju formulation): for Ns = 1,4,...,N/4:
//   v[r] = src[j + r*N/4];  v[r] *= cis(sign*2*pi*r*(j%Ns)/(4*Ns))
//   4-pt DFT with omega = sign*i;  dst[(j/Ns)*4*Ns + (j%Ns) + r*Ns] = X[r]

// First forward pass (Ns=1): real input, no twiddle.
__global__ __launch_bounds__(256) void fft_fwd_first(const float* __restrict__ nin,
                                                     float2* __restrict__ dst) {
    int t   = blockIdx.x * blockDim.x + threadIdx.x;   // BATCH*NQ threads
    int row = t >> 16;                                 // NQ == 2^16
    int j   = t & (NQ - 1);
    const float* s = nin + (size_t)row * NZ;
    float a = s[j], b = s[j + NQ], c = s[j + 2 * NQ], d = s[j + 3 * NQ];
    float t0 = a + c, t1 = a - c, t2 = b + d, t3 = b - d;   // all real
    float2* o = dst + (size_t)row * NZ + 4 * (size_t)j;
    o[0] = make_float2(t0 + t2, 0.0f);      // X0
    o[1] = make_float2(t1, -t3);            // X1 = t1 + (-i)*t3   (sign = -1)
    o[2] = make_float2(t0 - t2, 0.0f);      // X2
    o[3] = make_float2(t1,  t3);            // X3
}

// Generic pass, forward (sign=-1) or inverse (sign=+1).
__global__ __launch_bounds__(256) void fft_pass(const float2* __restrict__ src,
                                                float2* __restrict__ dst,
                                                int nsLog2, float sign) {
    int t   = blockIdx.x * blockDim.x + threadIdx.x;
    int row = t >> 16;
    int j   = t & (NQ - 1);
    int ns  = 1 << nsLog2;
    int jm  = j & (ns - 1);
    const float2* s = src + (size_t)row * NZ;
    float2 v0 = s[j], v1 = s[j + NQ], v2 = s[j + 2 * NQ], v3 = s[j + 3 * NQ];

    float ang = sign * (TWO_PI * 0.25f) * ((float)jm / (float)ns);
    float c1, s1, c2, s2, c3, s3;
    sincosf(ang, &s1, &c1);
    sincosf(2.0f * ang, &s2, &c2);
    sincosf(3.0f * ang, &s3, &c3);
    v1 = crot(v1, c1, s1);
    v2 = crot(v2, c2, s2);
    v3 = crot(v3, c3, s3);

    float2 t0 = cadd(v0, v2), t1 = csub(v0, v2);
    float2 t2 = cadd(v1, v3), t3 = csub(v1, v3);
    float2 w3 = make_float2(-sign * t3.y, sign * t3.x);    // (sign*i)*t3
    float2 X0 = cadd(t0, t2), X2 = csub(t0, t2);
    float2 X1 = cadd(t1, w3), X3 = csub(t1, w3);

    float2* d = dst + (size_t)row * NZ + ((size_t)(j >> nsLog2) << (nsLog2 + 2)) + jm;
    d[0]      = X0;
    d[ns]     = X1;
    d[2 * ns] = X2;
    d[3 * ns] = X3;
}

// First inverse pass (Ns=1) fused with spectral filter multiply.
// Even channel:  H[k] = w[min(k,N-k)]/N                (real, hermitian)
// Odd channel:   H[k] = +i*w[k]/N (k<=N/2), -i*w[N-k]/N (k>N/2)  (conj extension of i*w)
__global__ __launch_bounds__(256) void ifft_first_mul(const float2* __restrict__ X,
                                                      float2* __restrict__ dst,
                                                      const float* __restrict__ w,
                                                      int isOdd, float scale) {
    int t   = blockIdx.x * blockDim.x + threadIdx.x;
    int row = t >> 16;
    int j   = t & (NQ - 1);
    const float2* s = X + (size_t)row * NZ;

    float2 v[4];
#pragma unroll
    for (int r = 0; r < 4; ++r) {
        int k   = j + r * NQ;
        int idx = (k <= NZ / 2) ? k : (NZ - k);
        float sg = (k <= NZ / 2) ? 1.0f : -1.0f;
        float ww = w[idx] * scale;
        float2 x = s[k];
        if (isOdd) { float m = sg * ww; v[r] = make_float2(-x.y * m, x.x * m); } // (i*m)*x
        else       { v[r] = make_float2(x.x * ww, x.y * ww); }
    }
    // Ns=1 -> no twiddle; inverse butterfly (sign=+1)
    float2 t0 = cadd(v[0], v[2]), t1 = csub(v[0], v[2]);
    float2 t2 = cadd(v[1], v[3]), t3 = csub(v[1], v[3]);
    float2 w3 = make_float2(-t3.y, t3.x);                  // (+i)*t3
    float2* o = dst + (size_t)row * NZ + 4 * (size_t)j;
    o[0] = cadd(t0, t2);
    o[1] = cadd(t1, w3);
    o[2] = csub(t0, t2);
    o[3] = csub(t1, w3);
}

// Last inverse pass (Ns=N/4) fused with real-part extraction -> conv field.
__global__ __launch_bounds__(256) void ifft_last_real(const float2* __restrict__ src,
                                                      float* __restrict__ dst) {
    int t   = blockIdx.x * blockDim.x + threadIdx.x;
    int row = t >> 16;
    int j   = t & (NQ - 1);
    const float2* s = src + (size_t)row * NZ;
    float2 v0 = s[j], v1 = s[j + NQ], v2 = s[j + 2 * NQ], v3 = s[j + 3 * NQ];

    float ang = (TWO_PI * 0.25f) * ((float)j / (float)NQ);  // sign=+1
    float c1, s1, c2, s2, c3, s3;
    sincosf(ang, &s1, &c1);
    sincosf(2.0f * ang, &s2, &c2);
    sincosf(3.0f * ang, &s3, &c3);
    v1 = crot(v1, c1, s1);
    v2 = crot(v2, c2, s2);
    v3 = crot(v3, c3, s3);

    float2 t0 = cadd(v0, v2), t1 = csub(v0, v2);
    float2 t2 = cadd(v1, v3), t3 = csub(v1, v3);
    float2 w3 = make_float2(-t3.y, t3.x);
    float* d = dst + (size_t)row * NZ + j;
    d[0]      = t0.x + t2.x;
    d[NQ]     = t1.x + w3.x;
    d[2 * NQ] = t0.x - t2.x;
    d[3 * NQ] = t1.x - w3.x;
}

// ================= Pointwise MLP + reduction: fp32 WMMA 16x16x4 =================
__device__ __forceinline__ v8f wmma4(v2f a, v2f b, v8f c) {
    // D = A(16x4,f32) * B(4x16,f32) + C(16x16,f32)
    return __builtin_amdgcn_wmma_f32_16x16x4_f32(false, a, false, b, (short)0, c, false, false);
}
__device__ __forceinline__ v8f splat8(float s) {
    v8f v;
#pragma unroll
    for (int i = 0; i < 8; ++i) v[i] = s;
    return v;
}
__device__ __forceinline__ void softplus8(v8f& x) {
#pragma unroll
    for (int i = 0; i < 8; ++i) {
        float v = x[i];
        x[i] = fmaxf(v, 0.0f) + log1pf(expf(-fabsf(v)));   // stable softplus
    }
}

#define TILES_PER_B (NZ / 16)     // 16384
#define CONV_CS     (BATCH * NZ)  // channel stride in conv buffer

__global__ __launch_bounds__(256) void mlp_reduce(const float* __restrict__ conv,
                                                  const float* __restrict__ W1, const float* __restrict__ b1,
                                                  const float* __restrict__ W2, const float* __restrict__ b2,
                                                  const float* __restrict__ W3, const float* __restrict__ b3,
                                                  float* __restrict__ part, int totalWaves) {
    __shared__ __align__(16) float smem[8 * 768];           // per-wave: S(16x16) + H(16x32)
    const int lane = threadIdx.x & 31;
    const int wib  = threadIdx.x >> 5;
    const int wid  = (blockIdx.x * blockDim.x + threadIdx.x) >> 5;
    float* S = smem + wib * 768;                            // 256 floats, row stride 16
    float* H = S + 256;                                     // 512 floats, row stride 32
    const int hi = lane >> 4, lo = lane & 15;

    // ---- hoist all B-fragments (weights) into registers; ISA B layout:
    //      VGPR r, lane l -> K = r + 2*(l/16) (+4*ktile), N = l%16 (+16*ntile)
    v2f bw1[2][3], bw2[2][8], bw3[8];
#pragma unroll
    for (int n = 0; n < 2; ++n)
#pragma unroll
        for (int t = 0; t < 3; ++t) {
            int k0 = 4 * t + 2 * hi, c = 16 * n + lo;
            bw1[n][t] = (v2f){W1[k0 * 32 + c], W1[(k0 + 1) * 32 + c]};
        }
#pragma unroll
    for (int n = 0; n < 2; ++n)
#pragma unroll
        for (int t = 0; t < 8; ++t) {
            int k0 = 4 * t + 2 * hi, c = 16 * n + lo;
            bw2[n][t] = (v2f){W2[k0 * 32 + c], W2[(k0 + 1) * 32 + c]};
        }
#pragma unroll
    for (int t = 0; t < 8; ++t) {                           // W3 padded 12 -> 16 cols with 0
        int k0 = 4 * t + 2 * hi;
        float a = (lo < 12) ? W3[k0 * 12 + lo] : 0.0f;
        float b = (lo < 12) ? W3[(k0 + 1) * 12 + lo] : 0.0f;
        bw3[t] = (v2f){a, b};
    }
    const float bias1[2] = {b1[lo], b1[16 + lo]};
    const float bias2[2] = {b2[lo], b2[16 + lo]};
    const float bias3    = (lo < 12) ? b3[lo] : 0.0f;

    for (int b = 0; b < BATCH; ++b) {
        float accB = 0.0f;
        const float* cb = conv + (size_t)b * NZ;
        for (int tz = wid; tz < TILES_PER_B; tz += totalWaves) {   // uniform trip count
            // ---- build 12 scalar channels for 16 grid points, stage in LDS (A-side) ----
            int z = tz * 16 + lo;
            float e0 = cb[z];
            float e1 = cb[z + 1 * CONV_CS];
            float o0 = cb[z + 2 * CONV_CS];
            float o1 = cb[z + 3 * CONV_CS];
            float o2 = cb[z + 4 * CONV_CS];
            float o3 = cb[z + 5 * CONV_CS];
            float sv[12] = {e0, e1,
                            o0 * o0, o0 * o1, o0 * o2, o0 * o3,
                            o1 * o1, o1 * o2, o1 * o3,
                            o2 * o2, o2 * o3, o3 * o3};     // triu pairs of 4 odd channels
            float* Sp = S + lo * 16;
#pragma unroll
            for (int c = 0; c < 12; ++c) Sp[c] = sv[c];
            Sp[12] = Sp[13] = Sp[14] = Sp[15] = 0.0f;
            __syncthreads();

            // ---- layer 1: (16x12)x(12x32), K = 3 steps of 4 ----
            v2f a1[3];
#pragma unroll
            for (int t = 0; t < 3; ++t)
                a1[t] = *(const v2f*)(S + lo * 16 + 4 * t + 2 * hi);   // A: K=r+2*hi
            v8f h0 = splat8(bias1[0]), h1v = splat8(bias1[1]);
#pragma unroll
            for (int t = 0; t < 3; ++t) h0 = wmma4(a1[t], bw1[0][t], h0);
#pragma unroll
            for (int t = 0; t < 3; ++t) h1v = wmma4(a1[t], bw1[1][t], h1v);
            softplus8(h0); softplus8(h1v);
#pragma unroll
            for (int v = 0; v < 8; ++v) {                   // D layout: M=v+8*hi, N=lo
                H[(v + 8 * hi) * 32 + lo]      = h0[v];
                H[(v + 8 * hi) * 32 + 16 + lo] = h1v[v];
            }
            __syncthreads();

            // ---- layer 2: (16x32)x(32x32), K = 8 steps ----
            v2f a2[8];
#pragma unroll
            for (int t = 0; t < 8; ++t)
                a2[t] = *(const v2f*)(H + lo * 32 + 4 * t + 2 * hi);
            v8f g0 = splat8(bias2[0]), g1 = splat8(bias2[1]);
#pragma unroll
            for (int t = 0; t < 8; ++t) g0 = wmma4(a2[t], bw2[0][t], g0);
#pragma unroll
            for (int t = 0; t < 8; ++t) g1 = wmma4(a2[t], bw2[1][t], g1);
            softplus8(g0); softplus8(g1);
            __syncthreads();                                // all reads of H done block-wide
#pragma unroll
            for (int v = 0; v < 8; ++v) {
                H[(v + 8 * hi) * 32 + lo]      = g0[v];
                H[(v + 8 * hi) * 32 + 16 + lo] = g1[v];
            }
            __syncthreads();

            // ---- layer 3: (16x32)x(32x16pad), then dot with scalars ----
            v2f a3[8];
#pragma unroll
            for (int t = 0; t < 8; ++t)
                a3[t] = *(const v2f*)(H + lo * 32 + 4 * t + 2 * hi);
            v8f f = splat8(bias3);
#pragma unroll
            for (int t = 0; t < 8; ++t) f = wmma4(a3[t], bw3[t], f);
            float p = 0.0f;
#pragma unroll
            for (int v = 0; v < 8; ++v)
                p += f[v] * S[(v + 8 * hi) * 16 + lo];      // channel lo, 8 points each lane
            accB += p;
            __syncthreads();
        }
        // wave-level tree reduction, deterministic per-wave partial
#pragma unroll
        for (int off = 16; off; off >>= 1) accB += __shfl_xor(accB, off, 32);
        if (lane == 0) part[b * totalWaves + wid] = accB;
    }
}

__global__ __launch_bounds__(256) void final_reduce(const float* __restrict__ part,
                                                    float* __restrict__ out, int perB) {
    __shared__ float red[256];
    int b = blockIdx.x;
    float s = 0.0f;
    for (int i = threadIdx.x; i < perB; i += 256) s += part[(size_t)b * perB + i];
    red[threadIdx.x] = s;
    __syncthreads();
    for (int st = 128; st; st >>= 1) {
        if (threadIdx.x < st) red[threadIdx.x] += red[threadIdx.x + st];
        __syncthreads();
    }
    if (threadIdx.x == 0) out[b] = DZ_F * red[0];
}

// ================================ host launcher =================================
extern "C" void kernel_launch(void* const* d_in, const int* in_sizes, int n_in,
                              void* d_out, int out_size, void* d_ws, size_t ws_size,
                              hipStream_t stream) {
    const float* n_ptr  = (const float*)d_in[0];
    const float* w_even = (const float*)d_in[1];
    const float* w_odd  = (const float*)d_in[2];
    const float* W1 = (const float*)d_in[3];
    const float* b1 = (const float*)d_in[4];
    const float* W2 = (const float*)d_in[5];
    const float* b2 = (const float*)d_in[6];
    const float* W3 = (const float*)d_in[7];
    const float* b3 = (const float*)d_in[8];
    float* out = (float*)d_out;

    const size_t cplxBytes = (size_t)BATCH * NZ * sizeof(float2);   // 16 MB each
    char* ws = (char*)d_ws;
    float2* bufA = (float2*)(ws);
    float2* bufB = (float2*)(ws + cplxBytes);
    float2* Xs   = (float2*)(ws + 2 * cplxBytes);
    float*  conv = (float*)(ws + 3 * cplxBytes);                    // 6*8*NZ floats (48 MB)
    float*  part = (float*)(ws + 3 * cplxBytes + (size_t)6 * BATCH * NZ * sizeof(float));

    const int mlpBlocks  = 512;
    const int totalWaves = mlpBlocks * 256 / 32;                    // 4096; divides 16384
    const size_t needed  = 3 * cplxBytes + (size_t)6 * BATCH * NZ * sizeof(float)
                         + (size_t)BATCH * totalWaves * sizeof(float);
    if (ws_size < needed) return;

    dim3 blk(256);
    dim3 grdF((BATCH * NQ) / 256);                                  // 2048 blocks per pass

    // ---- forward FFT: 9 radix-4 Stockham passes, final pass lands in Xs ----
    fft_fwd_first<<<grdF, blk, 0, stream>>>(n_ptr, bufA);
    float2* cur = bufA; float2* oth = bufB;
    for (int p = 1; p < NPASS; ++p) {
        float2* dst = (p == NPASS - 1) ? Xs : oth;
        fft_pass<<<grdF, blk, 0, stream>>>(cur, dst, 2 * p, -1.0f);
        float2* tmp = cur; cur = oth; oth = tmp;
    }

    // ---- 6 filtered inverse FFTs -> real conv fields ----
    for (int c = 0; c < 6; ++c) {
        const float* w = (c < 2) ? (w_even + (size_t)c * NG) : (w_odd + (size_t)(c - 2) * NG);
        ifft_first_mul<<<grdF, blk, 0, stream>>>(Xs, bufA, w, (c >= 2) ? 1 : 0, 1.0f / (float)NZ);
        cur = bufA; oth = bufB;
        for (int p = 1; p < NPASS - 1; ++p) {
            fft_pass<<<grdF, blk, 0, stream>>>(cur, oth, 2 * p, 1.0f);
            float2* tmp = cur; cur = oth; oth = tmp;
        }
        ifft_last_real<<<grdF, blk, 0, stream>>>(cur, conv + (size_t)c * BATCH * NZ);
    }

    // ---- fused scalars -> WMMA MLP -> deterministic reduction ----
    mlp_reduce<<<mlpBlocks, blk, 0, stream>>>(conv, W1, b1, W2, b2, W3, b3, part, totalWaves);
    final_reduce<<<BATCH, blk, 0, stream>>>(part, out, totalWaves);
}